// CropAndResize_56212531970148
// MI455X (gfx1250) — compile-verified
//
#include <hip/hip_runtime.h>

// crop_and_resize, f32, CDNA5 (gfx1250, wave32).
// Separable bilinear: per (box, channel) wave:
//   stage 1: gather + horizontal lerp -> H[k=0..27][x=0..13] in LDS
//   stage 2: out(16x16) = Ry(16x28) @ H(28x16) via 7x V_WMMA_F32_16X16X4_F32
// Ry rows have two nonzeros (1-ly, ly) at k=2y, 2y+1; invalid y rows -> 0,
// invalid x columns zeroed in H (reference's extrapolation_value = 0).

typedef __attribute__((ext_vector_type(2))) float v2f;
typedef __attribute__((ext_vector_type(8))) float v8f;

#define CROP   14
#define IMG_H  100
#define IMG_W  100
#define NCH    256
#define KPAD   28      // 2 * CROP source rows (y0/y1 pairs, duplicates OK)
#define NPAD   16      // N padded to WMMA width
#define WAVES  8       // waves per block

__global__ __launch_bounds__(256)
void crop_resize_wmma_kernel(const float* __restrict__ image,
                             const float* __restrict__ boxes,
                             const int*   __restrict__ box_ind,
                             float*       __restrict__ out)
{
    __shared__ float Hs[WAVES * KPAD * NPAD];   // 8 * 448 * 4B = 14336 B

    const int n    = blockIdx.x;               // box index
    const int wave = threadIdx.x >> 5;
    const int lane = threadIdx.x & 31;
    const int c    = blockIdx.y * WAVES + wave; // channel

    // Box parameters (uniform per block -> scalar loads)
    const float by1 = boxes[n * 4 + 0];
    const float bx1 = boxes[n * 4 + 1];
    const float by2 = boxes[n * 4 + 2];
    const float bx2 = boxes[n * 4 + 3];
    const int   bi  = box_ind[n];

    const float hs = (float)(IMG_H - 1);
    const float ws = (float)(IMG_W - 1);
    const float stepy = (by2 - by1) * hs * (1.0f / (CROP - 1));
    const float stepx = (bx2 - bx1) * ws * (1.0f / (CROP - 1));
    const float basey = by1 * hs;
    const float basex = bx1 * ws;

    const float* plane = image + ((size_t)bi * NCH + c) * (size_t)(IMG_H * IMG_W);
    float* Hw = &Hs[wave * KPAD * NPAD];

    // ---- Stage 1: fill H[k][x] (horizontal lerp of needed source rows) ----
    // 448 elements / 32 lanes = 14 uniform iterations (no divergence on the loop)
    #pragma unroll
    for (int it = 0; it < (KPAD * NPAD) / 32; ++it) {
        const int idx = it * 32 + lane;
        const int k   = idx >> 4;      // 0..27
        const int x   = idx & 15;      // 0..15 (>=14 is padding)
        float val = 0.0f;
        if (x < CROP) {
            const int   y  = k >> 1;
            const float fy = basey + (float)y * stepy;
            const float fx = basex + (float)x * stepx;
            const bool  vx = (fx >= 0.0f) && (fx <= ws);
            const float yc = fminf(fmaxf(fy, 0.0f), hs);
            const float xc = fminf(fmaxf(fx, 0.0f), ws);
            const int   r0 = (int)floorf(yc);
            const int   x0 = (int)floorf(xc);
            const float lx = xc - (float)x0;
            const int   r  = (k & 1) ? min(r0 + 1, IMG_H - 1) : r0;
            const int   xr = min(x0 + 1, IMG_W - 1);
            const float* row = plane + r * IMG_W;
            const float p0 = row[x0];
            const float p1 = row[xr];
            val = vx ? (p0 + (p1 - p0) * lx) : 0.0f;   // extrapolation = 0
        }
        Hw[idx] = val;
    }
    __syncthreads();

    // ---- Stage 2: D = Ry @ H via V_WMMA_F32_16X16X4_F32 (K = 28) ----
    // Per-lane vertical weights for row m = lane & 15.
    const int   m  = lane & 15;
    const float fy = basey + (float)m * stepy;
    const bool  vy = (fy >= 0.0f) && (fy <= hs) && (m < CROP);
    const float yc = fminf(fmaxf(fy, 0.0f), hs);
    const float ly = yc - floorf(yc);
    const float w0 = vy ? (1.0f - ly) : 0.0f;  // weight at k = 2m
    const float w1 = vy ? ly          : 0.0f;  // weight at k = 2m + 1

    // 16x4 f32 A layout: VGPR0 holds K=kbase (lanes 0-15) / K=kbase+2 (lanes 16-31),
    // VGPR1 holds K=kbase+1 / K=kbase+3. B mirrors with N across lanes.
    const int khalf = (lane >> 4) << 1;        // 0 or 2

    v8f acc = {};
    #pragma unroll
    for (int kbase = 0; kbase < KPAD; kbase += 4) {
        const int kA = kbase + khalf;
        v2f a, b;
        a.x = (kA       == 2 * m) ? w0 : ((kA       == 2 * m + 1) ? w1 : 0.0f);
        a.y = ((kA + 1) == 2 * m) ? w0 : (((kA + 1) == 2 * m + 1) ? w1 : 0.0f);
        b.x = Hw[kA       * NPAD + m];
        b.y = Hw[(kA + 1) * NPAD + m];
        acc = __builtin_amdgcn_wmma_f32_16x16x4_f32(
                  /*neg_a=*/false, a, /*neg_b=*/false, b,
                  /*c_mod=*/(short)0, acc, /*reuse_a=*/false, /*reuse_b=*/false);
    }

    // ---- Store D: lane -> column x = lane&15; rows y = (lane>=16 ? 8 : 0) + v ----
    const int x = lane & 15;
    if (x < CROP) {
        const int rbase = (lane >> 4) * 8;
        float* outp = out + ((size_t)n * NCH + c) * (size_t)(CROP * CROP);
        #pragma unroll
        for (int v = 0; v < 8; ++v) {
            const int yy = rbase + v;
            if (yy < CROP) outp[yy * CROP + x] = acc[v];
        }
    }
}

extern "C" void kernel_launch(void* const* d_in, const int* in_sizes, int n_in,
                              void* d_out, int out_size, void* d_ws, size_t ws_size,
                              hipStream_t stream)
{
    const float* image   = (const float*)d_in[0];
    const float* boxes   = (const float*)d_in[1];
    const int*   box_ind = (const int*)d_in[2];
    float*       out     = (float*)d_out;

    const int N = in_sizes[2];          // 512 boxes
    dim3 grid(N, NCH / WAVES);          // (512, 32)
    crop_resize_wmma_kernel<<<grid, 256, 0, stream>>>(image, boxes, box_ind, out);
}